// CauseEffectRepertoire_5007931867616
// MI455X (gfx1250) — compile-verified
//
#include <hip/hip_runtime.h>
#include <math.h>

// ---------------------------------------------------------------------------
// CauseEffectRepertoire phi on gfx1250 (MI455X), f32 WMMA path.
//   D=16, H=64, B=8, M=32767 masks, 2 directions.
//   Per mask: X (16x16) = [state*bits ; state*(1-bits)]  (8 rows each)
//   H = relu(X @ W1^T + b1)       : 16 x v_wmma_f32_16x16x4_f32
//   O = H @ W2^T + b2             : 16 x v_wmma_f32_16x16x4_f32
//   softmax rows -> mixture -> KL vs precomputed full -> min over masks.
// ---------------------------------------------------------------------------

#define ND    16
#define NH    64
#define NB    8
#define NMASK 32767
#define EPSF  1e-10f
#define BLOCKS_PER_DIR 64            // 8 waves each -> 512 waves per direction
#define WAVES_PER_DIR  (BLOCKS_PER_DIR * 8)

typedef __attribute__((ext_vector_type(2))) float v2f;
typedef __attribute__((ext_vector_type(8))) float v8f;

static __device__ __forceinline__ v8f wmma_f32x4(v2f a, v2f b, v8f c) {
  // D = A(16x4,f32) x B(4x16,f32) + C(16x16,f32)
  return __builtin_amdgcn_wmma_f32_16x16x4_f32(
      /*neg_a=*/false, a, /*neg_b=*/false, b,
      /*c_mod=*/(short)0, c, /*reuse_a=*/false, /*reuse_b=*/false);
}

// monotonic float -> int key (atomicMin-able); decode inverts.
static __device__ __forceinline__ int f2ord(float f) {
  int b = __float_as_int(f);
  return (b >= 0) ? b : (b ^ 0x7fffffff);
}
static __device__ __forceinline__ float ord2f(int k) {
  int b = (k >= 0) ? k : (k ^ 0x7fffffff);
  return __int_as_float(b);
}

// ---------------------------------------------------------------------------
// Kernel 1: init min-keys to +inf; compute full = softmax(MLP(state)) per
// (dir, b) -- 16 tiny serial evaluations.
// ---------------------------------------------------------------------------
__global__ void cer_init_kernel(
    const float* __restrict__ state,
    const float* __restrict__ cw1, const float* __restrict__ cb1,
    const float* __restrict__ cw2, const float* __restrict__ cb2,
    const float* __restrict__ ew1, const float* __restrict__ eb1,
    const float* __restrict__ ew2, const float* __restrict__ eb2,
    float* __restrict__ ws_full, int* __restrict__ ws_min) {
  int t = threadIdx.x;
  if (t < 16) ws_min[t] = f2ord(__builtin_inff());
  if (t < 16) {
    int dir = t >> 3, b = t & 7;
    const float* w1 = dir ? ew1 : cw1;
    const float* b1 = dir ? eb1 : cb1;
    const float* w2 = dir ? ew2 : cw2;
    const float* b2 = dir ? eb2 : cb2;
    float h[NH];
    for (int j = 0; j < NH; ++j) {
      float s = b1[j];
      for (int d = 0; d < ND; ++d) s += state[b * ND + d] * w1[j * ND + d];
      h[j] = s > 0.f ? s : 0.f;
    }
    float lg[ND];
    float mx = -3.4e38f;
    for (int n = 0; n < ND; ++n) {
      float s = b2[n];
      for (int j = 0; j < NH; ++j) s += h[j] * w2[n * NH + j];
      lg[n] = s;
      mx = fmaxf(mx, s);
    }
    float sum = 0.f;
    for (int n = 0; n < ND; ++n) { lg[n] = __expf(lg[n] - mx); sum += lg[n]; }
    for (int n = 0; n < ND; ++n)
      ws_full[dir * (NB * ND) + b * ND + n] = lg[n] / sum;
  }
}

// ---------------------------------------------------------------------------
// Kernel 2: main mask sweep.  blockIdx.x = dir + 2*blk.  One wave == one mask
// per loop trip; all 32 lanes always active (WMMA needs EXEC all-ones).
// ---------------------------------------------------------------------------
__global__ __launch_bounds__(256) void cer_phi_kernel(
    const float* __restrict__ state,
    const float* __restrict__ cw1, const float* __restrict__ cb1,
    const float* __restrict__ cw2, const float* __restrict__ cb2,
    const float* __restrict__ ew1, const float* __restrict__ eb1,
    const float* __restrict__ ew2, const float* __restrict__ eb2,
    const float* __restrict__ ws_full, int* __restrict__ ws_min) {
  const int dir = blockIdx.x & 1;
  const int blk = blockIdx.x >> 1;
  const float* w1 = dir ? ew1 : cw1;
  const float* b1 = dir ? eb1 : cb1;
  const float* w2 = dir ? ew2 : cw2;
  const float* b2 = dir ? eb2 : cb2;

  __shared__ float s_state[NB * ND];
  __shared__ float s_w1[NH * ND];
  __shared__ float s_b1[NH];
  __shared__ float s_w2[ND * NH];
  __shared__ float s_b2[ND];
  __shared__ float s_full[NB * ND];
  __shared__ float s_H[8][16 * 65];   // per-wave 16x64 H tile, stride 65 (bank-conflict pad)

  const int tid = threadIdx.x;
  for (int i = tid; i < NB * ND; i += 256) {
    s_state[i] = state[i];
    s_full[i]  = ws_full[dir * (NB * ND) + i];
  }
  for (int i = tid; i < NH * ND; i += 256) s_w1[i] = w1[i];
  for (int i = tid; i < ND * NH; i += 256) s_w2[i] = w2[i];
  for (int i = tid; i < NH; i += 256) s_b1[i] = b1[i];
  for (int i = tid; i < ND; i += 256) s_b2[i] = b2[i];
  __syncthreads();

  const int lane = tid & 31;
  const int wv   = tid >> 5;
  const int hi   = lane >> 4;        // half-wave select
  const int ln   = lane & 15;
  float* Hs = s_H[wv];

  // ---- mask-invariant B fragments, resident in VGPRs -------------------
  // Layer 1: B[k][n] = W1[n][k]  (4 K-chunks x 4 N-tiles)
  // B layout assumption (mirrors documented A layout): v0 = rows {k0, k0+2}
  // across lane halves, v1 = rows {k0+1, k0+3}; here packed as (k0, k0+1)
  // per half with k0 = 4*kk + 2*hi.
  v2f bf1[4][4];
#pragma unroll
  for (int kk = 0; kk < 4; ++kk)
#pragma unroll
    for (int nt = 0; nt < 4; ++nt) {
      int n = nt * 16 + ln;
      int k0 = kk * 4 + hi * 2;
      v2f t;
      t.x = s_w1[n * ND + k0];
      t.y = s_w1[n * ND + k0 + 1];
      bf1[kk][nt] = t;
    }
  // Layer 2: B[k][n] = W2[n][k]  (16 K-chunks, 1 N-tile)
  v2f bf2[16];
#pragma unroll
  for (int kk = 0; kk < 16; ++kk) {
    int k0 = kk * 4 + hi * 2;
    v2f t;
    t.x = s_w2[ln * NH + k0];
    t.y = s_w2[ln * NH + k0 + 1];
    bf2[kk] = t;
  }

  float b1v[4];
#pragma unroll
  for (int nt = 0; nt < 4; ++nt) b1v[nt] = s_b1[nt * 16 + ln];
  const float b2v = s_b2[ln];
  float fullv[8];
#pragma unroll
  for (int i = 0; i < 8; ++i) fullv[i] = s_full[i * ND + ln];  // full[b=i][n=ln]

  float minv[8];
#pragma unroll
  for (int i = 0; i < 8; ++i) minv[i] = __builtin_inff();

  const int row  = ln;                 // A-matrix row this lane owns (M = lane&15)
  const int bsel = row & 7;            // batch index
  const bool isA = row < 8;            // rows 0-7: state_a, rows 8-15: state_b

  const int wave = blk * 8 + wv;
  for (int m = 1 + wave; m <= NMASK; m += WAVES_PER_DIR) {
    // ---- build X A-fragments (16x16 over 4 K-chunks) -------------------
    v2f af[4];
#pragma unroll
    for (int kk = 0; kk < 4; ++kk) {
      int k0 = kk * 4 + hi * 2;
      float x0 = s_state[bsel * ND + k0];
      float x1 = s_state[bsel * ND + k0 + 1];
      float m0 = (float)((m >> k0) & 1);
      float m1 = (float)((m >> (k0 + 1)) & 1);
      if (!isA) { m0 = 1.f - m0; m1 = 1.f - m1; }
      v2f t; t.x = x0 * m0; t.y = x1 * m1;
      af[kk] = t;
    }

    // ---- layer 1: H = relu(X @ W1^T + b1), 16 WMMAs --------------------
#pragma unroll
    for (int nt = 0; nt < 4; ++nt) {
      v8f acc = {};
#pragma unroll
      for (int kk = 0; kk < 4; ++kk) acc = wmma_f32x4(af[kk], bf1[kk][nt], acc);
      float bb = b1v[nt];
#pragma unroll
      for (int i = 0; i < 8; ++i) {           // C layout: M = i + 8*hi, N = ln
        float h = acc[i] + bb;
        h = h > 0.f ? h : 0.f;
        Hs[(i + hi * 8) * 65 + nt * 16 + ln] = h;
      }
    }

    // ---- layer 2: O = H @ W2^T + b2, 16 WMMAs --------------------------
    v8f acc2 = {};
#pragma unroll
    for (int kk = 0; kk < 16; ++kk) {
      int k0 = kk * 4 + hi * 2;
      v2f a;
      a.x = Hs[row * 65 + k0];
      a.y = Hs[row * 65 + k0 + 1];
      acc2 = wmma_f32x4(a, bf2[kk], acc2);
    }

    // ---- softmax rows, mixture, KL, running min ------------------------
    // acc2[i] @ lane: O[i + 8*hi][ln] -> lanes 0-15 = rep_a[b=i], 16-31 = rep_b[b=i]
#pragma unroll
    for (int i = 0; i < 8; ++i) {
      float x = acc2[i] + b2v;
      float mx = x;
      mx = fmaxf(mx, __shfl_xor(mx, 1));
      mx = fmaxf(mx, __shfl_xor(mx, 2));
      mx = fmaxf(mx, __shfl_xor(mx, 4));
      mx = fmaxf(mx, __shfl_xor(mx, 8));
      float e = __expf(x - mx);
      float sm = e;
      sm += __shfl_xor(sm, 1);
      sm += __shfl_xor(sm, 2);
      sm += __shfl_xor(sm, 4);
      sm += __shfl_xor(sm, 8);
      float p = e / sm;
      float q = 0.5f * (p + __shfl_xor(p, 16));   // mix rep_a and rep_b
      float f = fullv[i];
      float t = f * __log2f(fmaxf(f, EPSF) / fmaxf(q, EPSF));
      t += __shfl_xor(t, 1);
      t += __shfl_xor(t, 2);
      t += __shfl_xor(t, 4);
      t += __shfl_xor(t, 8);
      minv[i] = fminf(minv[i], t);
    }
  }

  if (lane == 0) {
#pragma unroll
    for (int i = 0; i < 8; ++i)
      atomicMin(&ws_min[dir * 8 + i], f2ord(minv[i]));
  }
}

// ---------------------------------------------------------------------------
// Kernel 3: decode min keys -> output (2, 8) floats.
// ---------------------------------------------------------------------------
__global__ void cer_fin_kernel(const int* __restrict__ ws_min,
                               float* __restrict__ out) {
  int t = threadIdx.x;
  if (t < 16) out[t] = ord2f(ws_min[t]);
}

extern "C" void kernel_launch(void* const* d_in, const int* in_sizes, int n_in,
                              void* d_out, int out_size, void* d_ws, size_t ws_size,
                              hipStream_t stream) {
  const float* state = (const float*)d_in[0];
  const float* cw1 = (const float*)d_in[1];
  const float* cb1 = (const float*)d_in[2];
  const float* cw2 = (const float*)d_in[3];
  const float* cb2 = (const float*)d_in[4];
  const float* ew1 = (const float*)d_in[5];
  const float* eb1 = (const float*)d_in[6];
  const float* ew2 = (const float*)d_in[7];
  const float* eb2 = (const float*)d_in[8];
  float* out = (float*)d_out;

  int*   ws_min  = (int*)d_ws;            // 16 ints
  float* ws_full = (float*)d_ws + 16;     // 2*8*16 floats

  cer_init_kernel<<<1, 64, 0, stream>>>(state, cw1, cb1, cw2, cb2,
                                        ew1, eb1, ew2, eb2, ws_full, ws_min);
  cer_phi_kernel<<<2 * BLOCKS_PER_DIR, 256, 0, stream>>>(
      state, cw1, cb1, cw2, cb2, ew1, eb1, ew2, eb2, ws_full, ws_min);
  cer_fin_kernel<<<1, 16, 0, stream>>>(ws_min, out);
}